// Link_Embedding_2422361555499
// MI455X (gfx1250) — compile-verified
//
#include <hip/hip_runtime.h>

// Link embedding gather: out[e] = concat(X[src[e]], X[dst[e]])
// X: [100000, 128] f32, indices: [320000, 2] int, out: [320000, 256] f32.
//
// Pure memory-bound gather (0 FLOPs). Strategy for MI455X (gfx1250, wave32):
//  - one wave per edge; edge id forced wave-uniform with readfirstlane so the
//    index pair loads as a single scalar s_load_b64
//  - 128 floats per row == 32 lanes x float4 -> one global_load_b128 per lane
//    per row half (perfectly coalesced 512B/wave transactions)
//  - output is write-once / never re-read (328 MB stream): use non-temporal
//    stores so we don't evict the 51 MB feature table from the 192 MB L2
//    (feature rows are re-read ~6.4x on average and should stay L2-resident)

typedef float v4f __attribute__((ext_vector_type(4)));

#define D_FEAT 128
#define WAVES_PER_BLOCK 8
#define BLOCK_THREADS (WAVES_PER_BLOCK * 32)

__global__ __launch_bounds__(BLOCK_THREADS) void link_embed_gather_kernel(
    const float* __restrict__ X,
    const int* __restrict__ indices,   // [n_edges, 2] (src, dst)
    float* __restrict__ out,           // [n_edges, 2*D_FEAT]
    int n_edges) {
  const int lane = threadIdx.x & 31;
  // Wave-uniform edge id: readfirstlane makes uniformity provable so the
  // index load below can be selected as a scalar (SMEM) load.
  const int wave_in_block = __builtin_amdgcn_readfirstlane(threadIdx.x >> 5);
  const int edge = blockIdx.x * WAVES_PER_BLOCK + wave_in_block;
  if (edge >= n_edges) return;   // uniform branch across the wave

  // Both node ids for this edge in one 8-byte scalar load.
  const int2 pair = *(const int2*)(indices + 2 * (size_t)edge);
  const long long src = (long long)pair.x;
  const long long dst = (long long)pair.y;

  // One float4 per lane covers a full 128-float row per wave (32*16B = 512B).
  const v4f* srcRow = (const v4f*)(X + src * D_FEAT);
  const v4f* dstRow = (const v4f*)(X + dst * D_FEAT);
  v4f a = srcRow[lane];
  v4f b = dstRow[lane];

  // Concatenated output row: [src features | dst features], streamed NT.
  v4f* orow = (v4f*)(out + (size_t)edge * (2 * D_FEAT));
  __builtin_nontemporal_store(a, orow + lane);
  __builtin_nontemporal_store(b, orow + 32 + lane);
}

extern "C" void kernel_launch(void* const* d_in, const int* in_sizes, int n_in,
                              void* d_out, int out_size, void* d_ws, size_t ws_size,
                              hipStream_t stream) {
  const float* X = (const float*)d_in[0];
  const int* indices = (const int*)d_in[1];
  float* out = (float*)d_out;

  // in_sizes[1] is the flat element count of the [n_edges, 2] index array.
  const int n_edges = in_sizes[1] / 2;

  const int blocks = (n_edges + WAVES_PER_BLOCK - 1) / WAVES_PER_BLOCK;
  link_embed_gather_kernel<<<dim3(blocks), dim3(BLOCK_THREADS), 0, stream>>>(
      X, indices, out, n_edges);
}